// TransGenerator_58420145160699
// MI455X (gfx1250) — compile-verified
//
#include <hip/hip_runtime.h>

typedef __attribute__((ext_vector_type(2))) float v2f;
typedef __attribute__((ext_vector_type(8))) float v8f;

static constexpr int BB   = 2;        // batch
static constexpr int CC   = 8;        // channels
static constexpr long PVOL = 64L * 64 * 64;        // 262144

// ---------------------------------------------------------------------------
// Stage A: conv3d 3x3x3 (C=8->8) + bias, then 2x2x2 avg-pool.
// GEMM view: D[16x16] += A[16x4] * B[4x16], K padded to 288 = 8ci*3kz*3ky*4kx
// (kx=3 is a zero weight column so each lane's two B values are LDS-adjacent
// and every inner-loop LDS address folds to an immediate offset).
// Block = 128 threads (4 waves); block covers 2z x 2y x 64x full-res conv
// = 16 tiles of (16 x-positions x 16 M-rows); each wave does 4 tiles.
// ---------------------------------------------------------------------------
__global__ __launch_bounds__(128)
void convpool_kernel(const float* __restrict__ xin,
                     const float* __restrict__ wconv,
                     const float* __restrict__ bconv,
                     float* __restrict__ DnOut,   // [B][C][64][64][64]
                     float* __restrict__ SOut)    // [B][64][64][64]
{
    __shared__ __align__(16) float inLds[8 * 4 * 4 * 66 + 2]; // halo + pad
    __shared__ __align__(16) float Wlds[16 * 288];            // padded weights
    __shared__ float convLds[2 * 2 * 64 * 8];                 // conv out
    __shared__ float bLds[8];
    __shared__ float sLds[32];

    const int tid = threadIdx.x;
    const int bid = blockIdx.x;
    const int xblock = bid & 1;
    const int yp = (bid >> 1) & 63;
    const int zp = (bid >> 7) & 63;
    const int b  = bid >> 13;

    __builtin_prefetch(wconv, 0, 0);

    // --- padded weight matrix [16][288]: rows >=8 and kx==3 are zero ---
    for (int i = tid; i < 16 * 288; i += 128) {
        const int co = i / 288;
        const int kk = i - co * 288;
        const int kx = kk & 3;
        const int t  = kk >> 2;
        const int ci = t / 9;
        const int rr = t - ci * 9;
        const int kz = rr / 3;
        const int ky = rr - kz * 3;
        float v = 0.0f;
        if (co < 8 && kx < 3)
            v = wconv[co * 216 + ci * 27 + kz * 9 + ky * 3 + kx];
        Wlds[i] = v;
    }
    if (tid < 8)  bLds[tid] = bconv[tid];
    if (tid < 32) sLds[tid] = 0.0f;
    if (tid == 0) { inLds[8448] = 0.0f; inLds[8449] = 0.0f; }

    // --- input halo with zero padding: [c][zz 0..3][yy 0..3][xx 0..65]
    // one thread per (c,zz,yy) row: 128 rows of 66 contiguous elements ---
    {
        const int yy = tid & 3;
        const int zz = (tid >> 2) & 3;
        const int c  = tid >> 4;
        const int zg = 2 * zp - 1 + zz;
        const int yg = 2 * yp - 1 + yy;
        const int xg0 = 64 * xblock - 1;
        const bool rowOK = (unsigned)zg < 128u && (unsigned)yg < 128u;
        const float* src =
            xin + (((size_t)(b * 8 + c) * 128 + (rowOK ? zg : 0)) * 128 +
                   (rowOK ? yg : 0)) * 128;
        __builtin_prefetch(src, 0, 0);
        float* dst = &inLds[tid * 66];
        for (int xx = 0; xx < 66; ++xx) {
            const int xg = xg0 + xx;
            dst[xx] = (rowOK && (unsigned)xg < 128u) ? src[xg] : 0.0f;
        }
    }
    __syncthreads();

    const int lane = tid & 31;
    const int wave = tid >> 5;
    const int half = lane >> 4;   // upper 16 lanes
    const int mrow = lane & 15;   // A: M-row, B: N-col

    // per-lane invariant bases (all inner-loop offsets become immediates)
    const float* Arow = &Wlds[mrow * 288 + 2 * half];

    for (int tt = wave * 4; tt < wave * 4 + 4; ++tt) {
        const int dz = tt >> 3;
        const int dy = (tt >> 2) & 1;
        const int xbase = (tt & 3) * 16;

        const float* Brow =
            &inLds[dz * 264 + dy * 66 + xbase + mrow + 2 * half];

        // C init: lanes 0..15 hold M=r (real channels) -> seed with bias
        v8f acc;
        #pragma unroll
        for (int r = 0; r < 8; ++r) acc[r] = (half == 0) ? bLds[r] : 0.0f;

        // 72 fully-unrolled K-steps: ds_load_b64(A) + 2x ds_load_b32(B) + wmma
        #pragma unroll
        for (int t = 0; t < 72; ++t) {
            const int ci = t / 9;
            const int rr = t - ci * 9;
            const int kz = rr / 3;
            const int ky = rr - kz * 3;
            const int rowofs = ci * 1056 + kz * 264 + ky * 66; // constant

            const v2f a = *(const v2f*)(Arow + 4 * t);
            v2f bf;
            bf.x = Brow[rowofs];
            bf.y = Brow[rowofs + 1];

            acc = __builtin_amdgcn_wmma_f32_16x16x4_f32(
                false, a, false, bf, (short)0, acc, false, false);
        }

        // lanes 0..15 hold (M=r, N=mrow) -> conv value for channel r
        if (half == 0) {
            const int basec = ((dz * 2 + dy) * 64 + xbase + mrow) * 8;
            #pragma unroll
            for (int r = 0; r < 8; ++r) convLds[basec + r] = acc[r];
        }
    }
    __syncthreads();

    // --- 2x2x2 average pool: 32 pooled-x * 8 channels = 256 outputs ---
    for (int it = 0; it < 2; ++it) {
        const int o = tid + it * 128;
        const int xp = o >> 3;
        const int co = o & 7;
        float sum = 0.0f;
        #pragma unroll
        for (int dz2 = 0; dz2 < 2; ++dz2)
            #pragma unroll
            for (int dy2 = 0; dy2 < 2; ++dy2)
                #pragma unroll
                for (int dx2 = 0; dx2 < 2; ++dx2)
                    sum += convLds[((dz2 * 2 + dy2) * 64 + 2 * xp + dx2) * 8 + co];
        sum *= 0.125f;
        DnOut[(((size_t)(b * 8 + co) * 64 + zp) * 64 + yp) * 64 + (xblock * 32 + xp)] = sum;
        atomicAdd(&sLds[xp], sum);   // ds_add_f32: channel sum for S
    }
    __syncthreads();
    if (tid < 32)
        SOut[(((size_t)b * 64 + zp) * 64 + yp) * 64 + (xblock * 32 + tid)] = sLds[tid];
}

// ---------------------------------------------------------------------------
// Stage B: Gaussian 5x5x5 of S and the three coordinate-weighted fields.
// w_gauss is channel-broadcast, so the reference's Gaussian conv equals the
// filter applied to the channel-sum S; the grid field is channel-independent
// and equals grids.mean(axis=1) exactly. Writes [B][64^3][3] output region.
// ---------------------------------------------------------------------------
__global__ __launch_bounds__(256)
void gauss_kernel(const float* __restrict__ S,
                  const float* __restrict__ wgauss,
                  float* __restrict__ gout)
{
    __shared__ float kerLds[125];
    const int tid = threadIdx.x;
    if (tid < 125) kerLds[tid] = wgauss[tid];   // (co=0,ci=0) 5x5x5 slab
    __syncthreads();

    const int idx = blockIdx.x * 256 + tid;     // [0, B*PVOL)
    const int vox = idx & (int)(PVOL - 1);
    const int b   = idx >> 18;
    const int z = vox >> 12, y = (vox >> 6) & 63, x = vox & 63;
    const float inv63 = 1.0f / 63.0f;
    const float* Sb = S + (size_t)b * PVOL;

    float ps = 0.0f, f0 = 0.0f, f1 = 0.0f, f2 = 0.0f;
    for (int dz = 0; dz < 5; ++dz) {
        const int zs = z + dz - 2;
        if ((unsigned)zs >= 64u) continue;
        for (int dy = 0; dy < 5; ++dy) {
            const int ys = y + dy - 2;
            if ((unsigned)ys >= 64u) continue;
            for (int dx = 0; dx < 5; ++dx) {
                const int xs = x + dx - 2;
                if ((unsigned)xs >= 64u) continue;
                const float t = kerLds[(dz * 5 + dy) * 5 + dx] *
                                Sb[((zs * 64) + ys) * 64 + xs];
                ps += t;
                f0 += t * (zs * inv63);   // P axis 0 = h (z) coord
                f1 += t * (ys * inv63);   // P axis 1 = w (y) coord
                f2 += t * (xs * inv63);   // P axis 2 = d (x) coord
            }
        }
    }
    const float denom = ps + 1e-6f;
    const float r0 = fminf(fmaxf(f0 / denom * 2.0f - 1.0f, -1.0f), 1.0f);
    const float r1 = fminf(fmaxf(f1 / denom * 2.0f - 1.0f, -1.0f), 1.0f);
    const float r2 = fminf(fmaxf(f2 / denom * 2.0f - 1.0f, -1.0f), 1.0f);
    float* g = gout + (size_t)idx * 3;
    g[0] = r0; g[1] = r1; g[2] = r2;
}

// ---------------------------------------------------------------------------
// Stage C: trilinear grid_sample (zeros padding, align_corners=False).
// One thread per (b, pooled voxel); grid shared by all 8 channels.
// grid component 0 -> last axis (x), 1 -> y, 2 -> z (torch convention).
// ---------------------------------------------------------------------------
__global__ __launch_bounds__(256)
void sample_kernel(const float* __restrict__ xin,
                   const float* __restrict__ grids,  // [B][PVOL][3]
                   float* __restrict__ yout)         // [B][C][PVOL]
{
    const int idx = blockIdx.x * 256 + threadIdx.x;  // [0, B*PVOL)
    const int vox = idx & (int)(PVOL - 1);
    const int b   = idx >> 18;

    const float* g = grids + (size_t)idx * 3;
    const float ix = ((g[0] + 1.0f) * 128.0f - 1.0f) * 0.5f;
    const float iy = ((g[1] + 1.0f) * 128.0f - 1.0f) * 0.5f;
    const float iz = ((g[2] + 1.0f) * 128.0f - 1.0f) * 0.5f;
    const float x0f = floorf(ix), y0f = floorf(iy), z0f = floorf(iz);
    const float wx = ix - x0f, wy = iy - y0f, wz = iz - z0f;
    const int x0 = (int)x0f, y0 = (int)y0f, z0 = (int)z0f;

    const float wxs[2] = {1.0f - wx, wx};
    const float wys[2] = {1.0f - wy, wy};
    const float wzs[2] = {1.0f - wz, wz};
    const int   xs[2] = {x0, x0 + 1}, ys[2] = {y0, y0 + 1}, zs[2] = {z0, z0 + 1};

    size_t cidx[8]; float cwt[8];
    int n = 0;
    #pragma unroll
    for (int a = 0; a < 2; ++a)
        #pragma unroll
        for (int e = 0; e < 2; ++e)
            #pragma unroll
            for (int c2 = 0; c2 < 2; ++c2) {
                const int zc = zs[a], yc = ys[e], xc = xs[c2];
                const bool ok = (unsigned)zc < 128u && (unsigned)yc < 128u &&
                                (unsigned)xc < 128u;
                const int zcl = min(max(zc, 0), 127);
                const int ycl = min(max(yc, 0), 127);
                const int xcl = min(max(xc, 0), 127);
                cidx[n] = ((size_t)zcl * 128 + ycl) * 128 + xcl;
                cwt[n]  = ok ? (wzs[a] * wys[e] * wxs[c2]) : 0.0f;
                ++n;
            }

    #pragma unroll
    for (int c = 0; c < 8; ++c) {
        const float* vol = xin + ((size_t)(b * 8 + c) << 21);
        float acc = 0.0f;
        #pragma unroll
        for (int k = 0; k < 8; ++k) acc += vol[cidx[k]] * cwt[k];
        yout[(size_t)(b * 8 + c) * PVOL + vox] = acc;
    }
}

// ---------------------------------------------------------------------------
extern "C" void kernel_launch(void* const* d_in, const int* in_sizes, int n_in,
                              void* d_out, int out_size, void* d_ws, size_t ws_size,
                              hipStream_t stream) {
    (void)in_sizes; (void)n_in; (void)out_size; (void)ws_size;
    const float* x      = (const float*)d_in[0];
    const float* wconv  = (const float*)d_in[1];
    const float* bconv  = (const float*)d_in[2];
    const float* wgauss = (const float*)d_in[3];

    float* out   = (float*)d_out;
    float* yOut  = out;                                   // [2][8][64^3]
    float* DnOut = out + (size_t)BB * CC * PVOL;          // [2][8][64^3]
    float* gOut  = out + (size_t)2 * BB * CC * PVOL;      // [2][64^3][3]
    float* S     = (float*)d_ws;                          // [2][64^3]

    // Stage A: grid = B * 64 zp * 64 yp * 2 xblocks = 16384 blocks
    convpool_kernel<<<16384, 128, 0, stream>>>(x, wconv, bconv, DnOut, S);
    // Stage B + C: one thread per (b, pooled voxel)
    gauss_kernel<<<(int)((BB * PVOL) / 256), 256, 0, stream>>>(S, wgauss, gOut);
    sample_kernel<<<(int)((BB * PVOL) / 256), 256, 0, stream>>>(x, gOut, yOut);
}